// AFGCN_5119601017047
// MI455X (gfx1250) — compile-verified
//
#include <hip/hip_runtime.h>

typedef __attribute__((ext_vector_type(2))) float v2f;
typedef __attribute__((ext_vector_type(8))) float v8f;

#define N_NODES 100000
#define N_EDGES 800000
#define D 64
#define L 4

// ---------------------------------------------------------------- zero agg
__global__ void afgcn_zero(float* __restrict__ p, int n) {
    int i = blockIdx.x * blockDim.x + threadIdx.x;
    if (i < n) p[i] = 0.0f;
}

// ---------------------------------------------------------------- scatter-add
// 16 lanes per edge; each lane: one float4 gather of h[src], 4 fp32 atomics
// into agg[dst]. agg (25.6MB) lives in L2 (192MB), atomics stay on-chip.
__global__ void afgcn_scatter(const float* __restrict__ h,
                              const long long* __restrict__ src,
                              const long long* __restrict__ dst,
                              float* __restrict__ agg) {
    int gid   = blockIdx.x * blockDim.x + threadIdx.x;
    int edge  = gid >> 4;
    int lane4 = (gid & 15) * 4;
    if (edge >= N_EDGES) return;
    int s = (int)src[edge];
    int d = (int)dst[edge];
    const float4 v = *(const float4*)(h + (size_t)s * D + lane4);
    float* a = agg + (size_t)d * D + lane4;
    unsafeAtomicAdd(a + 0, v.x);
    unsafeAtomicAdd(a + 1, v.y);
    unsafeAtomicAdd(a + 2, v.z);
    unsafeAtomicAdd(a + 3, v.w);
}

// ---------------------------------------------------------------- fused layer GEMM
// hNext = relu(agg @ Wrel + brel + h @ Wroot) + x0
// One 16x16 output tile per wave32 via V_WMMA_F32_16X16X4_F32, K=64 -> 16 steps
// per input matrix, both matmuls accumulate into the same C fragment.
__global__ void afgcn_gemm(const float* __restrict__ agg,
                           const float* __restrict__ h,
                           const float* __restrict__ Wrel,   // 64x64 row-major
                           const float* __restrict__ brel,   // 64
                           const float* __restrict__ Wroot,  // 64x64 row-major
                           const float* __restrict__ x0,     // residual (N x 64)
                           float* __restrict__ out) {
    const int lane    = threadIdx.x & 31;
    const int wave    = threadIdx.x >> 5;
    const int rowTile = blockIdx.x * 2 + (wave >> 2);  // 32 rows per block
    const int colTile = wave & 3;                      // 4 col tiles cover D=64
    const int rowBase = rowTile * 16;
    const int colBase = colTile * 16;
    const int half    = lane >> 4;   // 0: K pair {k,k+1}; 1: {k+2,k+3}
    const int r16     = lane & 15;

    const float* aggRow = agg + (size_t)(rowBase + r16) * D;
    const float* hRow   = h   + (size_t)(rowBase + r16) * D;

    v8f c = {};
#pragma unroll
    for (int k = 0; k < 16; ++k) {
        const int kk = k * 4 + half * 2;
        // A fragment (16x4 f32 layout): lane holds row r16, cols kk, kk+1
        v2f a1 = *(const v2f*)(aggRow + kk);
        v2f b1;
        b1.x = Wrel[(size_t)kk * D + colBase + r16];
        b1.y = Wrel[(size_t)(kk + 1) * D + colBase + r16];
        c = __builtin_amdgcn_wmma_f32_16x16x4_f32(
                false, a1, false, b1, (short)0, c, false, false);

        v2f a2 = *(const v2f*)(hRow + kk);
        v2f b2;
        b2.x = Wroot[(size_t)kk * D + colBase + r16];
        b2.y = Wroot[(size_t)(kk + 1) * D + colBase + r16];
        c = __builtin_amdgcn_wmma_f32_16x16x4_f32(
                false, a2, false, b2, (short)0, c, false, false);
    }

    // epilogue: C/D layout = VGPR i, lanes 0-15 -> row rowBase+i,
    // lanes 16-31 -> row rowBase+8+i, col = colBase + (lane&15)
    const float bias = brel[colBase + r16];
#pragma unroll
    for (int i = 0; i < 8; ++i) {
        const int row = rowBase + half * 8 + i;
        const int col = colBase + r16;
        float v = c[i] + bias;
        v = v > 0.0f ? v : 0.0f;
        v += x0[(size_t)row * D + col];
        out[(size_t)row * D + col] = v;
    }
}

// ---------------------------------------------------------------- final matvec
__global__ void afgcn_fc(const float* __restrict__ h,
                         const float* __restrict__ w,
                         const float* __restrict__ b,
                         float* __restrict__ out) {
    int i = blockIdx.x * blockDim.x + threadIdx.x;
    if (i >= N_NODES) return;
    const float4* hp = (const float4*)(h + (size_t)i * D);
    const float4* wp = (const float4*)w;
    float s = 0.0f;
#pragma unroll
    for (int j = 0; j < D / 4; ++j) {
        float4 hv = hp[j];
        float4 wv = wp[j];
        s += hv.x * wv.x + hv.y * wv.y + hv.z * wv.z + hv.w * wv.w;
    }
    out[i] = s + b[0];
}

extern "C" void kernel_launch(void* const* d_in, const int* in_sizes, int n_in,
                              void* d_out, int out_size, void* d_ws, size_t ws_size,
                              hipStream_t stream) {
    const float*     x     = (const float*)d_in[0];
    const long long* ei    = (const long long*)d_in[1];   // int64 (2, E)
    const float*     Wrel  = (const float*)d_in[2];       // (L, 64, 64)
    const float*     brel  = (const float*)d_in[3];       // (L, 64)
    const float*     Wroot = (const float*)d_in[4];       // (L, 64, 64)
    const float*     fcw   = (const float*)d_in[5];       // (64, 1)
    const float*     fcb   = (const float*)d_in[6];       // (1,)
    float*           out   = (float*)d_out;

    const long long* src = ei;            // row 0
    const long long* dst = ei + N_EDGES;  // row 1

    float* agg = (float*)d_ws;
    float* hA  = agg + (size_t)N_NODES * D;
    float* hB  = hA  + (size_t)N_NODES * D;

    const float* hcur = x;
    float* bufs[2] = {hA, hB};
    int cur = 0;

    for (int l = 0; l < L; ++l) {
        afgcn_zero<<<(N_NODES * D + 255) / 256, 256, 0, stream>>>(agg, N_NODES * D);
        afgcn_scatter<<<(N_EDGES * 16) / 256, 256, 0, stream>>>(hcur, src, dst, agg);
        float* hnext = bufs[cur];
        afgcn_gemm<<<N_NODES / 32, 256, 0, stream>>>(
            agg, hcur,
            Wrel + (size_t)l * D * D,
            brel + (size_t)l * D,
            Wroot + (size_t)l * D * D,
            x, hnext);
        hcur = hnext;
        cur ^= 1;
    }
    afgcn_fc<<<(N_NODES + 255) / 256, 256, 0, stream>>>(hcur, fcw, fcb, out);
}